// CAELoss_16509854286417
// MI455X (gfx1250) — compile-verified
//
#include <hip/hip_runtime.h>
#include <hip/hip_bf16.h>
#include <math.h>

typedef float v2f __attribute__((ext_vector_type(2)));
typedef float v8f __attribute__((ext_vector_type(8)));

#define C_CLS   10
#define L_DIM   128
#define D_IN_M  0.1f

// ws layout (floats):
//  [0,1280)      normalized centers (10x128)
//  [1280,1296)   ||c||^2 padded to 16
//  [1296]        loss_orth
//  [1300,1364)   TC block partials (64)
//  [1400,1464)   OUT block partials (64)
//  [1536,2560)   MSE block partials (1024)
#define WS_CN    1280
#define WS_ORTH  1296
#define WS_TC    1300
#define WS_OUT   1400
#define WS_MSE   1536

// ---------------- Kernel A: center normalization, ||c||^2, orthogonality loss ----
__global__ __launch_bounds__(128) void prep_centers(const float* __restrict__ center_arr,
                                                    float* __restrict__ ws) {
    __shared__ float sc[C_CLS * L_DIM];
    __shared__ float snorm[C_CLS];
    __shared__ float sgram[C_CLS * C_CLS];
    const int tid = threadIdx.x;

    for (int i = tid; i < C_CLS * L_DIM; i += 128) sc[i] = center_arr[i];
    __syncthreads();

    if (tid < C_CLS) {
        float s = 0.0f;
        for (int l = 0; l < L_DIM; ++l) { float v = sc[tid * L_DIM + l]; s += v * v; }
        snorm[tid] = sqrtf(s);
    }
    __syncthreads();

    for (int i = tid; i < C_CLS * L_DIM; i += 128) {
        float v = sc[i] / snorm[i >> 7];
        sc[i] = v;
        ws[i] = v;
    }
    __syncthreads();

    if (tid < 16) {
        float s = 0.0f;
        if (tid < C_CLS)
            for (int l = 0; l < L_DIM; ++l) { float v = sc[tid * L_DIM + l]; s += v * v; }
        ws[WS_CN + tid] = s;   // padded classes -> 0 (unused)
    }
    if (tid < C_CLS * C_CLS) {
        int i = tid / C_CLS, j = tid % C_CLS;
        float d = 0.0f;
        for (int l = 0; l < L_DIM; ++l) d += sc[i * L_DIM + l] * sc[j * L_DIM + l];
        float e = d - (i == j ? 1.0f : 0.0f);
        sgram[tid] = e * e;
    }
    __syncthreads();
    if (tid == 0) {
        float s = 0.0f;
        for (int i = 0; i < C_CLS * C_CLS; ++i) s += sgram[i];
        ws[WS_ORTH] = sqrtf(s);
    }
}

// ---------------- Kernel B: MSE block partials (float4 grid-stride) -------------
__global__ __launch_bounds__(256) void mse_partial(const float4* __restrict__ x,
                                                   const float4* __restrict__ xh,
                                                   float* __restrict__ ws, int n4) {
    __shared__ float red[256];
    const int tid = threadIdx.x;
    size_t idx = (size_t)blockIdx.x * 256 + tid;
    const size_t stride = (size_t)gridDim.x * 256;
    float s = 0.0f;
    for (size_t i = idx; i < (size_t)n4; i += stride) {
        float4 a = x[i], b = xh[i];
        float dx = a.x - b.x, dy = a.y - b.y, dz = a.z - b.z, dw = a.w - b.w;
        s += dx * dx + dy * dy + dz * dz + dw * dw;
    }
    red[tid] = s;
    __syncthreads();
    for (int o = 128; o > 0; o >>= 1) {
        if (tid < o) red[tid] += red[tid + o];
        __syncthreads();
    }
    if (tid == 0) ws[WS_MSE + blockIdx.x] = red[0];
}

// ---------------- Kernel C: triplet-center loss (WMMA) + outlier loss ----------
// 128 threads = 4 waves; each wave owns 16 rows; block covers 64 rows.
__global__ __launch_bounds__(128) void tc_out_kernel(const float* __restrict__ z_in,
                                                     const float* __restrict__ z_out,
                                                     const int* __restrict__ target,
                                                     float* __restrict__ ws) {
    __shared__ float sC[16 * L_DIM];        // centers padded to 16 classes
    __shared__ float sZ[4 * 16 * L_DIM];    // per-wave z_in tile
    __shared__ float sD[4 * 16 * 16];       // per-wave dot tile
    __shared__ float sZn[4 * 16];           // per-wave ||z||^2
    __shared__ float sCn[16];               // ||c||^2
    __shared__ float red[128];

    const int tid  = threadIdx.x;
    const int wave = tid >> 5;
    const int lane = tid & 31;
    const int baseRow = blockIdx.x * 64;
    const int wRow = baseRow + wave * 16;

    // padded centers + norms -> LDS
    for (int i = tid; i < 16 * L_DIM; i += 128)
        sC[i] = ((i >> 7) < C_CLS) ? ws[i] : 0.0f;
    if (tid < 16) sCn[tid] = ws[WS_CN + tid];

    // stage this wave's 16x128 z_in tile (coalesced)
    for (int i = lane; i < 16 * L_DIM; i += 32)
        sZ[wave * 2048 + i] = z_in[(size_t)wRow * L_DIM + i];
    __syncthreads();

    // WMMA f32 16x16x4 over K=128 (32 chunks).
    // A layout: lane M = lane&15; VGPR0/1 hold K = 4k+2h, 4k+2h+1 (h = lane>>4)
    // B layout: lane N = lane&15; same K striping (B[K][N] = centers[N][K])
    const int M = lane & 15;
    const int h = lane >> 4;
    const float* zrow = &sZ[wave * 2048 + M * L_DIM];
    const float* crow = &sC[M * L_DIM];   // class index == lane&15
    v8f acc = {};
    float zn = 0.0f;
    for (int k = 0; k < 32; ++k) {
        const int K = 4 * k + 2 * h;
        v2f a; a.x = zrow[K]; a.y = zrow[K + 1];
        v2f b; b.x = crow[K]; b.y = crow[K + 1];
        zn += a.x * a.x + a.y * a.y;
        acc = __builtin_amdgcn_wmma_f32_16x16x4_f32(false, a, false, b,
                                                    (short)0, acc, false, false);
    }
    // combine the two K-halves of ||z||^2 (lane l and l^16 cover K%4 in {0,1}/{2,3})
    zn += __shfl_xor(zn, 16, 32);
    if (h == 0) sZn[wave * 16 + M] = zn;

    // D layout: VGPR r -> row (r + 8h), col (lane&15)
#pragma unroll
    for (int r = 0; r < 8; ++r)
        sD[wave * 256 + (r + 8 * h) * 16 + M] = acc[r];
    __syncthreads();

    // hinge(pos + margin - min_neg), lanes 0..15 of each wave: one row each
    float tc = 0.0f;
    if (lane < 16) {
        const int row = wRow + lane;
        const int t = target[row];
        const float z2 = sZn[wave * 16 + lane];
        float pos = 0.0f, neg = INFINITY;
        for (int c = 0; c < C_CLS; ++c) {
            float d2 = z2 + sCn[c] - 2.0f * sD[wave * 256 + lane * 16 + c];
            float d = sqrtf(fmaxf(d2, 0.0f));
            if (c == t) pos = d; else neg = fminf(neg, d);
        }
        tc = fmaxf(pos + D_IN_M - neg, 0.0f);
    }

    // outlier loss: 64 rows per block, 2 threads/row (64 elements each)
    {
        const int r = tid >> 1;
        const int off = (tid & 1) * 64;
        const float* zo = &z_out[((size_t)baseRow + r) * L_DIM + off];
        float os = 0.0f;
        for (int l = 0; l < 64; ++l) { float v = zo[l]; os += v * v; }
        os += __shfl_xor(os, 1, 32);
        float oc = (tid & 1) ? 0.0f : fmaxf(1.0f - sqrtf(os), 0.0f);
        red[tid] = oc;
    }
    __syncthreads();
    for (int o = 64; o > 0; o >>= 1) {
        if (tid < o) red[tid] += red[tid + o];
        __syncthreads();
    }
    if (tid == 0) ws[WS_OUT + blockIdx.x] = red[0];
    __syncthreads();

    red[tid] = tc;
    __syncthreads();
    for (int o = 64; o > 0; o >>= 1) {
        if (tid < o) red[tid] += red[tid + o];
        __syncthreads();
    }
    if (tid == 0) ws[WS_TC + blockIdx.x] = red[0];
}

// ---------------- Kernel D: combine everything ---------------------------------
__global__ __launch_bounds__(256) void finalize(const float* __restrict__ ws,
                                                float* __restrict__ out,
                                                int nMse, int nTc, int nTotal, int B) {
    __shared__ float red[256];
    const int tid = threadIdx.x;
    const float invN = 1.0f / (float)nTotal;
    const float invB = 1.0f / (float)B;
    float s = 0.0f;
    for (int i = tid; i < nMse; i += 256) s += ws[WS_MSE + i] * invN;
    if (tid < nTc) s += (ws[WS_TC + tid] + ws[WS_OUT + tid]) * invB;
    red[tid] = s;
    __syncthreads();
    for (int o = 128; o > 0; o >>= 1) {
        if (tid < o) red[tid] += red[tid + o];
        __syncthreads();
    }
    if (tid == 0) out[0] = red[0] + ws[WS_ORTH];
}

extern "C" void kernel_launch(void* const* d_in, const int* in_sizes, int n_in,
                              void* d_out, int out_size, void* d_ws, size_t ws_size,
                              hipStream_t stream) {
    const float* x       = (const float*)d_in[0];
    const float* x_hat   = (const float*)d_in[1];
    const int*   target  = (const int*)d_in[2];
    const float* z_in    = (const float*)d_in[3];
    const float* z_out   = (const float*)d_in[4];
    const float* centers = (const float*)d_in[5];
    float* out = (float*)d_out;
    float* ws  = (float*)d_ws;

    const int nTotal = in_sizes[0];          // 4096*3072
    const int B      = in_sizes[2];          // 4096
    const int mseBlocks = 1024;
    const int tcBlocks  = B / 64;            // 64

    prep_centers<<<1, 128, 0, stream>>>(centers, ws);
    mse_partial<<<mseBlocks, 256, 0, stream>>>((const float4*)x, (const float4*)x_hat,
                                               ws, nTotal / 4);
    tc_out_kernel<<<tcBlocks, 128, 0, stream>>>(z_in, z_out, target, ws);
    finalize<<<1, 256, 0, stream>>>(ws, out, mseBlocks, tcBlocks, nTotal, B);
}